// NMS_52132313038913
// MI455X (gfx1250) — compile-verified
//
#include <hip/hip_runtime.h>
#include <stdint.h>

// Batched greedy NMS for MI455X (gfx1250).
// One workgroup per batch; everything batch-local lives in LDS (WGP has 320KB).
//   Phase 1: build 64-bit stable sort keys (score-monotonic bits || ~index)
//   Phase 2: in-LDS bitonic sort, descending  -> JAX stable argsort(-scores)
//            (pair-index formulation: exactly N/2 compare-exchanges per stage)
//   Phase 3: extract sorted original indices
//   Phase 4: gather boxes in sorted order via per-lane async global->LDS DMA
//            (global_load_async_to_lds_b128 + s_wait_asynccnt, CDNA5-only path)
//   Phase 5: canonicalize corners into SoA + areas
//   Phase 6: greedy suppression, <= max_out kept. The NEXT cursor is computed
//            inside the suppression sweep via LDS atomicMin (ds_min), so the
//            serial chain per kept box is just 2 barriers + 1 LDS read.
//   Phase 7: pad remaining output rows with the batch's box 0 (TF padded-NMS)

#define NBOX   2048
#define BLOCK  256
#define EPT    (NBOX / BLOCK)   // 8 elements per thread
#define IOU_THR 0.7f

// LDS layout (bytes)
#define OFF_KEY   0        // u64[2048] = 16384; reused after sort:
#define OFF_SIDX  0        //   u32[2048] sorted original indices @ [0, 8192)
#define OFF_AREA  8192     //   f32[2048] areas                  @ [8192, 16384)
#define OFF_Y1    16384    // f32[2048] canonical y1
#define OFF_X1    24576    // f32[2048] canonical x1
#define OFF_Y2    32768    // f32[2048] canonical y2
#define OFF_X2    40960    // f32[2048] canonical x2
#define OFF_RAW   49152    // f32[2048*4] raw AoS boxes (async DMA target) = 32768
#define OFF_SUPP  81920    // u8[2048] suppression flags
#define OFF_CTL   83968    // int[4]: [0]=cursor, [1]=kept, [2]=next-min
#define SMEM_BYTES 84000

__global__ __launch_bounds__(BLOCK)
void nms_gfx1250_kernel(const float* __restrict__ rois,
                        const float* __restrict__ scores,
                        float* __restrict__ out,
                        int max_out)
{
    extern __shared__ unsigned char smem[];
    uint64_t*      s_key  = (uint64_t*)(smem + OFF_KEY);
    uint32_t*      s_sidx = (uint32_t*)(smem + OFF_SIDX);
    float*         s_area = (float*)(smem + OFF_AREA);
    float*         s_y1   = (float*)(smem + OFF_Y1);
    float*         s_x1   = (float*)(smem + OFF_X1);
    float*         s_y2   = (float*)(smem + OFF_Y2);
    float*         s_x2   = (float*)(smem + OFF_X2);
    float*         s_raw  = (float*)(smem + OFF_RAW);
    unsigned char* s_supp = (unsigned char*)(smem + OFF_SUPP);
    int*           s_ctl  = (int*)(smem + OFF_CTL);

    const int tid = threadIdx.x;
    const int b   = blockIdx.x;
    const size_t base = (size_t)b * NBOX;   // element row base for this batch

    // ---- Phase 1: sort keys. Monotonic float->u32 map; low word ~t gives
    // stable tie-break (smaller original index sorts first under descending).
    for (int t = tid; t < NBOX; t += BLOCK) {
        float sc = scores[base + t];
        uint32_t u = __float_as_uint(sc);
        u = (u & 0x80000000u) ? ~u : (u | 0x80000000u);
        s_key[t] = ((uint64_t)u << 32) | (uint64_t)(~(uint32_t)t);
    }
    __syncthreads();

    // ---- Phase 2: bitonic sort, descending. Pair-index form: p in [0, N/2)
    // maps to the unique exchange pair (i1, i1|j), no idle iterations.
    for (unsigned k = 2; k <= NBOX; k <<= 1) {
        for (unsigned j = k >> 1; j > 0; j >>= 1) {
            for (unsigned p = tid; p < NBOX / 2; p += BLOCK) {
                unsigned lo = p & (j - 1);
                unsigned i1 = ((p ^ lo) << 1) | lo;
                unsigned i2 = i1 | j;
                uint64_t a = s_key[i1];
                uint64_t c = s_key[i2];
                bool desc = ((i1 & k) == 0);
                if (desc ? (a < c) : (a > c)) {
                    s_key[i1] = c;
                    s_key[i2] = a;
                }
            }
            __syncthreads();
        }
    }

    // ---- Phase 3: extract sorted original indices (register-stage to avoid
    // overlap hazard: s_sidx aliases the low half of s_key).
    uint32_t myidx[EPT];
    for (int r = 0; r < EPT; ++r)
        myidx[r] = ~(uint32_t)s_key[tid + r * BLOCK];
    __syncthreads();
    for (int r = 0; r < EPT; ++r)
        s_sidx[tid + r * BLOCK] = myidx[r];
    __syncthreads();

    // ---- Phase 4: gather boxes in sorted order straight into LDS with the
    // CDNA5 async DMA path (per-lane 16B gather, no VGPR round trip).
    for (int t = tid; t < NBOX; t += BLOCK) {   // 2048/256: all lanes active
        uint32_t oi  = s_sidx[t];
        uint64_t ga  = (uint64_t)(uintptr_t)(rois + (base + oi) * 4);
        uint32_t loff = (uint32_t)(uintptr_t)(&s_raw[t * 4]);
        asm volatile("global_load_async_to_lds_b128 %0, %1, off"
                     :: "v"(loff), "v"(ga) : "memory");
    }
    asm volatile("s_wait_asynccnt 0x0" ::: "memory");
    __syncthreads();

    // ---- Phase 5: canonicalize corners (TF behavior) into SoA + area.
    for (int t = tid; t < NBOX; t += BLOCK) {
        float b0 = s_raw[t * 4 + 0];
        float b1 = s_raw[t * 4 + 1];
        float b2 = s_raw[t * 4 + 2];
        float b3 = s_raw[t * 4 + 3];
        float y1 = fminf(b0, b2), y2 = fmaxf(b0, b2);
        float x1 = fminf(b1, b3), x2 = fmaxf(b1, b3);
        s_y1[t] = y1; s_x1[t] = x1; s_y2[t] = y2; s_x2[t] = x2;
        s_area[t] = (y2 - y1) * (x2 - x1);
        s_supp[t] = 0;
    }
    if (tid == 0) s_ctl[2] = 0;   // first cursor: sorted box 0 (unsuppressed)
    __syncthreads();

    // ---- Phase 6: greedy NMS. Cursor for the next round is produced by the
    // sweep itself (min surviving index via ds_min atomic) -> no serial scan.
    int k_local = 0;   // meaningful on tid 0 only
    int count;
    while (1) {
        if (tid == 0) {
            s_ctl[0] = s_ctl[2];   // next unsuppressed sorted index (or NBOX)
            s_ctl[1] = k_local;
            s_ctl[2] = NBOX;       // reset min-reduction register
        }
        __syncthreads();                       // B1: control visible, supp stable
        const int i = s_ctl[0];
        const int k = s_ctl[1];
        if (i >= NBOX || k >= max_out) { count = k; break; }  // uniform

        // Emit kept box k: ORIGINAL (uncanonicalized) coordinates.
        if (tid < 4) {
            uint32_t oi = s_sidx[i];
            out[((size_t)b * max_out + k) * 4 + tid] = rois[(base + oi) * 4 + tid];
        }

        const float iy1 = s_y1[i], ix1 = s_x1[i];
        const float iy2 = s_y2[i], ix2 = s_x2[i];
        const float ia  = s_area[i];
        int localnext = NBOX;                  // first survivor in my stride
        for (int jj = i + 1 + tid; jj < NBOX; jj += BLOCK) {
            if (!s_supp[jj]) {
                float yy1 = fmaxf(iy1, s_y1[jj]);
                float yy2 = fminf(iy2, s_y2[jj]);
                float xx1 = fmaxf(ix1, s_x1[jj]);
                float xx2 = fminf(ix2, s_x2[jj]);
                float ih = yy2 - yy1; ih = ih > 0.0f ? ih : 0.0f;
                float iw = xx2 - xx1; iw = iw > 0.0f ? iw : 0.0f;
                float inter = ih * iw;
                float uni = ia + s_area[jj] - inter;
                float iou = (uni > 0.0f) ? (inter / uni) : 0.0f;
                if (iou > IOU_THR) {
                    s_supp[jj] = 1;
                } else if (localnext == NBOX) {
                    localnext = jj;            // jj ascends: first = thread min
                }
            }
        }
        if (localnext < NBOX) atomicMin(&s_ctl[2], localnext);
        if (tid == 0) k_local = k + 1;
        __syncthreads();                       // B2: supp + min writes complete
    }

    // ---- Phase 7: pad remaining rows with the batch's box 0 (original coords).
    const float p0 = rois[base * 4 + 0];
    const float p1 = rois[base * 4 + 1];
    const float p2 = rois[base * 4 + 2];
    const float p3 = rois[base * 4 + 3];
    for (int r = count + tid; r < max_out; r += BLOCK) {
        size_t o = ((size_t)b * max_out + r) * 4;
        out[o + 0] = p0; out[o + 1] = p1; out[o + 2] = p2; out[o + 3] = p3;
    }
}

extern "C" void kernel_launch(void* const* d_in, const int* in_sizes, int n_in,
                              void* d_out, int out_size, void* d_ws, size_t ws_size,
                              hipStream_t stream) {
    (void)n_in; (void)d_ws; (void)ws_size;
    const float* rois   = (const float*)d_in[0];   // [B, 2048, 4] f32
    const float* scores = (const float*)d_in[1];   // [B, 2048, 1] f32
    // d_in[2] is max_output_size on device; derive shapes on host instead
    // (deterministic, capture-safe).
    int B = in_sizes[1] / NBOX;                    // 16
    if (B < 1) B = 1;
    int max_out = out_size / (B * 4);              // 300

    nms_gfx1250_kernel<<<dim3(B), dim3(BLOCK), SMEM_BYTES, stream>>>(
        rois, scores, (float*)d_out, max_out);
}